// WarpCTCLoss_77919296684354
// MI455X (gfx1250) — compile-verified
//
#include <hip/hip_runtime.h>
#include <stdint.h>

#define CTC_NEG_INF (-1e30f)

#ifndef __has_builtin
#define __has_builtin(x) 0
#endif

// ---------- CDNA5 async global->LDS (ASYNCcnt) helpers ----------
// ROCm clang-22 signature (from diagnostics):
//   __builtin_amdgcn_global_load_async_to_lds_b64(
//       global v2i* gsrc, local v2i* ldst, int imm_offset, int cpol)
typedef int ctc_v2i __attribute__((vector_size(8)));
typedef __attribute__((address_space(1))) ctc_v2i* ctc_g64_t;
typedef __attribute__((address_space(3))) ctc_v2i* ctc_l64_t;

#if __has_builtin(__builtin_amdgcn_global_load_async_to_lds_b64)
#define CTC_HAS_ASYNC_BUILTIN 1
#else
#define CTC_HAS_ASYNC_BUILTIN 0
#endif

#if __has_builtin(__builtin_amdgcn_s_wait_asynccnt)
#define CTC_WAIT_ASYNC(n) __builtin_amdgcn_s_wait_asynccnt(n)
#else
#define CTC_WAIT_ASYNC(n) asm volatile("s_wait_asynccnt %0" :: "i"(n) : "memory")
#endif

// Copy 8 bytes per lane from global to LDS asynchronously (GLOBAL_LOAD_ASYNC_TO_LDS_B64).
__device__ __forceinline__ void ctc_async_copy_b64(const float* gsrc, float* ldst) {
#if CTC_HAS_ASYNC_BUILTIN
  __builtin_amdgcn_global_load_async_to_lds_b64(
      (ctc_g64_t)gsrc, (ctc_l64_t)ldst, 0, 0);
#else
  unsigned loff = (unsigned)(unsigned long long)
      (__attribute__((address_space(3))) void*)ldst;
  asm volatile("global_load_async_to_lds_b64 %0, %1, off"
               :: "v"(loff), "v"(gsrc) : "memory");
#endif
}

__device__ __forceinline__ float ctc_lae2(float a, float b) {
  float m = fmaxf(a, b);
  if (m < -1e29f) return CTC_NEG_INF;
  return m + __logf(__expf(a - m) + __expf(b - m));
}

// ---------------- Kernel 1: per-row logsumexp over V=512 ----------------
// One wave32 per (b,t) row; 16 floats/lane via 4x float4 loads.
__global__ void __launch_bounds__(256)
ctc_lse_kernel(const float* __restrict__ logits, float* __restrict__ lse, int rows) {
  const int lane = threadIdx.x & 31;
  const int wid  = blockIdx.x * 8 + (threadIdx.x >> 5);
  if (wid >= rows) return;
  const float* row = logits + (size_t)wid * 512;

  float4 v0 = *(const float4*)(row +   0 + lane * 4);
  float4 v1 = *(const float4*)(row + 128 + lane * 4);
  float4 v2 = *(const float4*)(row + 256 + lane * 4);
  float4 v3 = *(const float4*)(row + 384 + lane * 4);

  float m = v0.x;
  m = fmaxf(m, v0.y); m = fmaxf(m, v0.z); m = fmaxf(m, v0.w);
  m = fmaxf(m, v1.x); m = fmaxf(m, v1.y); m = fmaxf(m, v1.z); m = fmaxf(m, v1.w);
  m = fmaxf(m, v2.x); m = fmaxf(m, v2.y); m = fmaxf(m, v2.z); m = fmaxf(m, v2.w);
  m = fmaxf(m, v3.x); m = fmaxf(m, v3.y); m = fmaxf(m, v3.z); m = fmaxf(m, v3.w);
#pragma unroll
  for (int off = 16; off > 0; off >>= 1) m = fmaxf(m, __shfl_xor(m, off, 32));

  float s = 0.0f;
  s += __expf(v0.x - m) + __expf(v0.y - m) + __expf(v0.z - m) + __expf(v0.w - m);
  s += __expf(v1.x - m) + __expf(v1.y - m) + __expf(v1.z - m) + __expf(v1.w - m);
  s += __expf(v2.x - m) + __expf(v2.y - m) + __expf(v2.z - m) + __expf(v2.w - m);
  s += __expf(v3.x - m) + __expf(v3.y - m) + __expf(v3.z - m) + __expf(v3.w - m);
#pragma unroll
  for (int off = 16; off > 0; off >>= 1) s += __shfl_xor(s, off, 32);

  if (lane == 0) lse[wid] = m + __logf(s);
}

// ---------------- Kernel 2: CTC alpha recursion ----------------
// One block (256 threads) per batch element b. Thread tid owns lattice state
// s = tid (tid 0 additionally owns s = 256). Logits rows are staged into
// triple-buffered LDS via async-to-LDS copies, prefetched one t ahead.
// Assumes V == 512, T <= 1024, S = 2U+1 <= 257.
__global__ void __launch_bounds__(256)
ctc_alpha_kernel(const float* __restrict__ logits,
                 const int*   __restrict__ targets,
                 const int*   __restrict__ loglens,
                 const int*   __restrict__ tlens,
                 const float* __restrict__ lse,
                 float*       __restrict__ out,
                 int T, int V, int U) {
  const int b   = blockIdx.x;
  const int tid = threadIdx.x;
  const int S   = 2 * U + 1;

  __shared__ __align__(16) float rowbuf[3][512];
  __shared__ __align__(16) float lseRow[1024];
  __shared__ __align__(16) float alphaS[2][264];  // [0..1] = -inf pad, s at idx s+2

  const float* lbase = logits + (size_t)b * T * V;
  const int len = loglens[b];

  for (int t = tid; t < T; t += 256) lseRow[t] = lse[b * T + t];

  // Extended label for s = tid (even s -> blank 0), and skip-transition flag.
  int  myExt  = 0;
  bool mySkip = false;
  if (tid < S && (tid & 1)) {
    const int k = tid >> 1;
    myExt = targets[b * U + k];
    if (tid >= 3) mySkip = (myExt != targets[b * U + k - 1]);
  }

  if (tid < 2) { alphaS[0][tid] = CTC_NEG_INF; alphaS[1][tid] = CTC_NEG_INF; }

  // Pipeline prologue: stage rows 0 and 1 (each thread copies 8B -> 2KB/row).
  ctc_async_copy_b64(lbase + (size_t)0 * V + tid * 2, &rowbuf[0][tid * 2]);
  if (len > 1) {
    ctc_async_copy_b64(lbase + (size_t)1 * V + tid * 2, &rowbuf[1][tid * 2]);
    CTC_WAIT_ASYNC(1);   // row 0 complete (async copies retire in order)
  } else {
    CTC_WAIT_ASYNC(0);
  }
  __syncthreads();

  // t = 0: only s=0 (blank) and s=1 (first label) reachable.
  {
    const float l0 = lseRow[0];
    if (tid < S) {
      const float lp = rowbuf[0][myExt] - l0;
      alphaS[0][tid + 2] = (tid <= 1) ? lp : CTC_NEG_INF;
    }
    if (tid == 0 && S > 256) alphaS[0][256 + 2] = CTC_NEG_INF;
  }

  for (int t = 1; t < len; ++t) {
    const int cb = t & 1, pb = cb ^ 1;
    if (t + 1 < len) {
      ctc_async_copy_b64(lbase + (size_t)(t + 1) * V + tid * 2,
                         &rowbuf[(t + 1) % 3][tid * 2]);
      CTC_WAIT_ASYNC(1);  // row t done; row t+1 may still be in flight
    } else {
      CTC_WAIT_ASYNC(0);
    }
    __syncthreads();  // row t visible to all waves; prev alpha writes visible

    const float lseT = lseRow[t];
    const float* rb  = rowbuf[t % 3];

    if (tid < S) {
      const float a0 = alphaS[pb][tid + 2];
      const float a1 = alphaS[pb][tid + 1];
      const float a2 = alphaS[pb][tid];
      float m = fmaxf(a0, a1);
      if (mySkip) m = fmaxf(m, a2);
      float acc;
      if (m < -1e29f) {
        acc = CTC_NEG_INF;
      } else {
        float ss = __expf(a0 - m) + __expf(a1 - m);
        if (mySkip) ss += __expf(a2 - m);
        acc = m + __logf(ss);
      }
      alphaS[cb][tid + 2] = acc + (rb[myExt] - lseT);
    }
    if (tid == 0 && S > 256) {  // s = 256: even (blank), no skip transition
      const float a0 = alphaS[pb][256 + 2];
      const float a1 = alphaS[pb][256 + 1];
      alphaS[cb][256 + 2] = ctc_lae2(a0, a1) + (rb[0] - lseT);
    }
  }

  __syncthreads();
  if (tid == 0) {
    const int fb   = (len - 1) & 1;
    const int last = 2 * tlens[b];
    const float aE = alphaS[fb][last + 2];
    const float aP = alphaS[fb][last + 1];
    out[b] = -ctc_lae2(aE, aP);
  }
}

extern "C" void kernel_launch(void* const* d_in, const int* in_sizes, int n_in,
                              void* d_out, int out_size, void* d_ws, size_t ws_size,
                              hipStream_t stream) {
  const float* logits  = (const float*)d_in[0];
  const int*   targets = (const int*)d_in[1];
  const int*   loglens = (const int*)d_in[2];
  const int*   tlens   = (const int*)d_in[3];
  float*       out     = (float*)d_out;

  const int B = in_sizes[2];
  const int U = in_sizes[1] / B;
  const int V = 512;                    // fixed by reference setup
  const int T = in_sizes[0] / (B * V);

  float* lse = (float*)d_ws;            // B*T floats of scratch (256 KB)

  const int rows = B * T;
  ctc_lse_kernel<<<(rows + 7) / 8, 256, 0, stream>>>(logits, lse, rows);
  ctc_alpha_kernel<<<B, 256, 0, stream>>>(logits, targets, loglens, tlens,
                                          lse, out, T, V, U);
  (void)n_in; (void)out_size; (void)ws_size;
}